// GatbertSelfAttention_35467839930962
// MI455X (gfx1250) — compile-verified
//
#include <hip/hip_runtime.h>
#include <hip/hip_bf16.h>
#include <stdint.h>

#define HIDDEN   256
#define HEADS    8
#define HEAD_DIM 32
#define NODES    8192
// Pad rows to 260 floats: row base = 1040 B (16B-aligned for async B128 LDS
// writes) and A-frag reads hit banks (4*row + k) % 64 -> conflict-free.
#define LDSP     (HIDDEN + 4)

typedef __attribute__((ext_vector_type(2))) float v2f;
typedef __attribute__((ext_vector_type(8))) float v8f;

// D = A(16x4, f32) * B(4x16, f32) + C(16x16, f32)  -- native fp32 WMMA
__device__ __forceinline__ v8f wmma_f32_k4(v2f a, v2f b, v8f c) {
    return __builtin_amdgcn_wmma_f32_16x16x4_f32(
        /*neg_a=*/false, a, /*neg_b=*/false, b,
        /*c_mod=*/(short)0, c, /*reuse_a=*/false, /*reuse_b=*/false);
}

// gfx1250 async DMA: global memory -> LDS, no VGPR staging, ASYNCcnt-tracked.
__device__ __forceinline__ void async_ld_lds_b128(void* lds_ptr, const void* gptr) {
    const uint32_t lds = (uint32_t)(uintptr_t)lds_ptr;   // flat[31:0] = LDS byte offset
    const uint64_t src = (uint64_t)(uintptr_t)gptr;
    asm volatile("global_load_async_to_lds_b128 %0, %1, off"
                 :: "v"(lds), "v"(src) : "memory");
}
__device__ __forceinline__ void wait_async0() {
    asm volatile("s_wait_asynccnt 0" ::: "memory");
}

__global__ void zero_f32_kernel(float* __restrict__ p, int n) {
    int i = blockIdx.x * blockDim.x + threadIdx.x;
    if (i < n) p[i] = 0.0f;
}

// ---------------------------------------------------------------------------
// Kernel 2: fused Q/K/V node projection.  One block = 16 node rows.
// 8 waves x 6 tiles = 48 tiles = 3 matrices x 16 column tiles.
// out[m][c] = sum_k X[m][k] * W[c][k] + b[c]   (x @ W.T + b)
// ---------------------------------------------------------------------------
__global__ __launch_bounds__(256)
void qkv_wmma_kernel(const float* __restrict__ X,
                     const float* __restrict__ Wq, const float* __restrict__ bq,
                     const float* __restrict__ Wk, const float* __restrict__ bk,
                     const float* __restrict__ Wv, const float* __restrict__ bv,
                     float* __restrict__ Q, float* __restrict__ K,
                     float* __restrict__ V) {
    __shared__ float xs[16][LDSP];
    const int tid = threadIdx.x;
    const int rowBase = blockIdx.x * 16;

    // Async-stage the 16x256 node slab into LDS (4 x B128 per thread).
    {
        const int r0  = tid >> 6;             // 0..3
        const int col = (tid & 63) * 4;       // 16B-aligned column
        #pragma unroll
        for (int p = 0; p < 4; ++p) {
            const int e = p * 4 + r0;
            async_ld_lds_b128(&xs[e][col], &X[(size_t)(rowBase + e) * HIDDEN + col]);
        }
    }
    wait_async0();
    __syncthreads();

    const int lane = tid & 31;
    const int wave = tid >> 5;
    const int mrow = lane & 15;   // A-matrix row (node within slab) / B,C column
    const int hi   = lane >> 4;   // lanes 16-31 carry K pair {2,3} and M rows 8-15

    for (int t = 0; t < 6; ++t) {
        const int idx   = wave * 6 + t;      // 0..47
        const int m     = idx >> 4;          // 0=Q, 1=K, 2=V
        const int cTile = idx & 15;
        const int col   = cTile * 16 + mrow;
        const float* W  = (m == 0) ? Wq : (m == 1) ? Wk : Wv;
        const float* bs = (m == 0) ? bq : (m == 1) ? bk : bv;
        float* O        = (m == 0) ? Q  : (m == 1) ? K  : V;

        v8f acc = {0.f, 0.f, 0.f, 0.f, 0.f, 0.f, 0.f, 0.f};
        #pragma unroll 8
        for (int k4 = 0; k4 < HIDDEN / 4; ++k4) {
            const int kb = k4 * 4 + hi * 2;
            v2f a; a.x = xs[mrow][kb];            a.y = xs[mrow][kb + 1];
            v2f b; b.x = W[col * HIDDEN + kb];    b.y = W[col * HIDDEN + kb + 1];
            acc = wmma_f32_k4(a, b, acc);
        }
        const float bb = bs[col];
        #pragma unroll
        for (int r = 0; r < 8; ++r)
            O[(size_t)(rowBase + r + hi * 8) * HIDDEN + col] = acc[r] + bb;
    }
}

// ---------------------------------------------------------------------------
// Kernel 3: fused edge kernel.  One block = 16 edges.
//   Ke = ev @ Wk.T + bk,  Ve = ev @ We.T + be   (WMMA, weights hot in L2)
//   logits = sum_d Qm*(Kn+Ke) / sqrt(32)  -> softmax over 8 heads
//   scatter atomicAdd( attn * (Vn+Ve) ) into out[b, s, :]
// Q/K/V tables (48 MB) live in L2, so the gathers/atomics stay on-chip.
// ---------------------------------------------------------------------------
__global__ __launch_bounds__(256)
void edge_attention_kernel(const float* __restrict__ EV,
                           const long long* __restrict__ EIDX,   // (3, E) int64
                           const float* __restrict__ Wk, const float* __restrict__ bk,
                           const float* __restrict__ We, const float* __restrict__ be,
                           const float* __restrict__ Q,
                           const float* __restrict__ K,
                           const float* __restrict__ V,
                           float* __restrict__ Out, int E) {
    __shared__ float ev[16][LDSP];        // edge_values tile
    __shared__ float ke[16][LDSP];        // Ke + bk
    __shared__ float ve[16][LDSP];        // Ve + be
    __shared__ int   rowS[16];            // b*N + s  (scatter target)
    __shared__ int   rowT[16];            // b*N + t  (gather source)
    __shared__ float logit[16][HEADS];
    __shared__ float attn[16][HEADS];

    const int tid   = threadIdx.x;
    const int eBase = blockIdx.x * 16;

    // Async-stage the 16x256 edge_values tile (the only HBM-streamed operand).
    {
        const int r0  = tid >> 6;
        const int col = (tid & 63) * 4;
        #pragma unroll
        for (int p = 0; p < 4; ++p) {
            const int e = p * 4 + r0;
            async_ld_lds_b128(&ev[e][col], &EV[(size_t)(eBase + e) * HIDDEN + col]);
        }
    }
    if (tid < 16) {
        const int e = eBase + tid;
        const int b = (int)EIDX[(size_t)0 * E + e];
        const int s = (int)EIDX[(size_t)1 * E + e];
        const int t = (int)EIDX[(size_t)2 * E + e];
        rowS[tid] = b * NODES + s;
        rowT[tid] = b * NODES + t;
    }
    if (eBase + 16 < E)   // stream-ahead on the next tile
        __builtin_prefetch(&EV[(size_t)(eBase + 16) * HIDDEN + tid], 0, 0);
    wait_async0();
    __syncthreads();

    // --- WMMA edge projections: each wave owns 2 column tiles of Ke and of Ve
    const int lane = tid & 31;
    const int wave = tid >> 5;
    const int mrow = lane & 15;
    const int hi   = lane >> 4;
    for (int p = 0; p < 4; ++p) {
        const int cTile = wave * 2 + (p & 1);
        const int which = p >> 1;                 // 0 = Ke, 1 = Ve
        const float* W  = which ? We : Wk;
        const float* bs = which ? be : bk;
        const int col   = cTile * 16 + mrow;

        v8f acc = {0.f, 0.f, 0.f, 0.f, 0.f, 0.f, 0.f, 0.f};
        #pragma unroll 8
        for (int k4 = 0; k4 < HIDDEN / 4; ++k4) {
            const int kb = k4 * 4 + hi * 2;
            v2f a; a.x = ev[mrow][kb];            a.y = ev[mrow][kb + 1];
            v2f b; b.x = W[col * HIDDEN + kb];    b.y = W[col * HIDDEN + kb + 1];
            acc = wmma_f32_k4(a, b, acc);
        }
        const float bb = bs[col];
        float (*dst)[LDSP] = which ? ve : ke;
        #pragma unroll
        for (int r = 0; r < 8; ++r)
            dst[r + hi * 8][col] = acc[r] + bb;
    }
    __syncthreads();

    // --- logits: one thread per (edge, head); Qm/Kn gathered straight from L2
    if (tid < 16 * HEADS) {
        const int e = tid >> 3;
        const int h = tid & 7;
        const size_t qb = (size_t)rowS[e] * HIDDEN + h * HEAD_DIM;
        const size_t kb = (size_t)rowT[e] * HIDDEN + h * HEAD_DIM;
        float acc = 0.0f;
        #pragma unroll
        for (int d = 0; d < HEAD_DIM; ++d)
            acc += Q[qb + d] * (K[kb + d] + ke[e][h * HEAD_DIM + d]);
        logit[e][h] = acc * 0.17677669529663687f;   // 1/sqrt(32)
    }
    __syncthreads();

    // --- softmax over the 8 heads (reference semantics): one thread per edge
    if (tid < 16) {
        float mx = logit[tid][0];
        #pragma unroll
        for (int h = 1; h < HEADS; ++h) mx = fmaxf(mx, logit[tid][h]);
        float ex[HEADS], sum = 0.0f;
        #pragma unroll
        for (int h = 0; h < HEADS; ++h) { ex[h] = __expf(logit[tid][h] - mx); sum += ex[h]; }
        const float inv = 1.0f / sum;
        #pragma unroll
        for (int h = 0; h < HEADS; ++h) attn[tid][h] = ex[h] * inv;
    }
    __syncthreads();

    // --- scatter: attn * (Vn + Ve) -> out[b, s, :]  (f32 atomics land in L2)
    const int h = tid >> 5;   // head of column `tid`
    for (int e = 0; e < 16; ++e) {
        const float vnv = V[(size_t)rowT[e] * HIDDEN + tid] + ve[e][tid];
        atomicAdd(&Out[(size_t)rowS[e] * HIDDEN + tid], attn[e][h] * vnv);
    }
}

extern "C" void kernel_launch(void* const* d_in, const int* in_sizes, int n_in,
                              void* d_out, int out_size, void* d_ws, size_t ws_size,
                              hipStream_t stream) {
    const float*     node_states = (const float*)d_in[0];
    const long long* eidx        = (const long long*)d_in[1];   // int64 (3, E)
    const float*     edge_values = (const float*)d_in[2];
    const float*     Wq = (const float*)d_in[3];
    const float*     bq = (const float*)d_in[4];
    const float*     Wk = (const float*)d_in[5];
    const float*     bk = (const float*)d_in[6];
    const float*     Wv = (const float*)d_in[7];
    const float*     bv = (const float*)d_in[8];
    const float*     We = (const float*)d_in[9];
    const float*     be = (const float*)d_in[10];
    float* out = (float*)d_out;

    const int rows = in_sizes[0] / HIDDEN;   // B*N = 16384
    const int E    = in_sizes[1] / 3;        // 262144

    // Workspace: Q, K, V node tables (3 x 16 MB; fits comfortably in ws & L2)
    float* Q = (float*)d_ws;
    float* K = Q + (size_t)rows * HIDDEN;
    float* V = K + (size_t)rows * HIDDEN;

    zero_f32_kernel<<<(out_size + 255) / 256, 256, 0, stream>>>(out, out_size);
    qkv_wmma_kernel<<<rows / 16, 256, 0, stream>>>(node_states, Wq, bq, Wk, bk,
                                                   Wv, bv, Q, K, V);
    edge_attention_kernel<<<E / 16, 256, 0, stream>>>(edge_values, eidx, Wk, bk,
                                                      We, be, Q, K, V, out, E);
}